// GaussianMask_45183055954095
// MI455X (gfx1250) — compile-verified
//
#include <hip/hip_runtime.h>
#include <math.h>

// ---------------------------------------------------------------------------
// GaussianMask for MI455X (gfx1250, wave32).
//
// Roofline: corr volume is 170 MB in + 170 MB out -> memory bound (~15.5us at
// 23.3 TB/s). The only matrix-shaped work is the 18432x256 @ 256x16 MLP, done
// with V_WMMA_F32_16X16X4_F32 (full f32 precision; tanh pre-activations are
// ~N(0,5.7) so low precision would perturb saturation). The big pass is a
// pure stream with a cheap per-element window factor, using non-temporal
// loads/stores so the 340MB stream doesn't pollute caches.
// ---------------------------------------------------------------------------

typedef __attribute__((ext_vector_type(2))) float v2f;
typedef __attribute__((ext_vector_type(4))) float f4;
typedef __attribute__((ext_vector_type(8))) float v8f;

#define Bn 8
#define Hn 48
#define Wn 48
#define Cn 256
#define HW (Hn * Wn)                    // 2304
#define NPIX (Bn * HW)                  // 18432
#define CORR_ELEMS ((size_t)NPIX * HW)  // 42,467,328
#define RADIUS_F 6.0f
#define EPS_F 1e-5f

// ---------------------------------------------------------------------------
// Stage 1: tt = tanh(x @ map_w^T + map_b) via WMMA, then the two 16->2 heads.
// One wave (32 threads) per 16-pixel tile.
//
// WMMA f32 16x16x4 operand layouts (ISA 7.12.2):
//   A (16x4):  lanes 0-15: M=lane,    v0=K0, v1=K1 ; lanes 16-31: K2,K3
//   B (4x16):  lanes 0-15: N=lane,    v0=K0, v1=K1 ; lanes 16-31: K2,K3
//   C/D 16x16: VGPR r: M=r (lanes 0-15), M=8+r (lanes 16-31), N=lane%16
// B[k,n] = map_w[n*256+k], so A and B use the same float2 addressing pattern.
// ---------------------------------------------------------------------------
__global__ void k_mlp_wmma(const float* __restrict__ x,
                           const float* __restrict__ map_w,
                           const float* __restrict__ map_b,
                           const float* __restrict__ mean_w,
                           const float* __restrict__ mean_b,
                           const float* __restrict__ cov_w,
                           const float* __restrict__ cov_b,
                           float* __restrict__ out_mean,  // (NPIX,2)
                           float* __restrict__ cvraw)     // (NPIX,2) ws
{
    __shared__ float tt[16 * 16];

    const int lane = threadIdx.x;      // 0..31, full wave (EXEC all 1s)
    const int row  = lane & 15;        // M for A / N for B
    const int kh   = lane >> 4;        // K half selector
    const int pixbase = blockIdx.x * 16;

    const float* ax = x + (size_t)(pixbase + row) * Cn + 2 * kh;
    const float* bw = map_w + (size_t)row * Cn + 2 * kh;

    v8f acc = {};
#pragma unroll 4
    for (int k0 = 0; k0 < Cn; k0 += 4) {
        v2f a, b;
        a.x = ax[k0]; a.y = ax[k0 + 1];
        b.x = bw[k0]; b.y = bw[k0 + 1];
        // 8 args: (neg_a, A, neg_b, B, c_mod, C, reuse_a, reuse_b)
        acc = __builtin_amdgcn_wmma_f32_16x16x4_f32(
            false, a, false, b, (short)0, acc, false, false);
    }

    const float bias = map_b[row];
#pragma unroll
    for (int r = 0; r < 8; ++r) {
        const int m = r + 8 * kh;                 // C/D row mapping
        tt[m * 16 + row] = tanhf(acc[r] + bias);  // N = row
    }
    __syncthreads();

    if (lane < 16) {
        const int pix = pixbase + lane;
        float m0 = mean_b[0], m1 = mean_b[1];
        float c0 = cov_b[0],  c1 = cov_b[1];
#pragma unroll
        for (int n = 0; n < 16; ++n) {
            const float t = tt[lane * 16 + n];
            m0 = fmaf(t, mean_w[n],      m0);   // mean_w[0,n]
            m1 = fmaf(t, mean_w[16 + n], m1);   // mean_w[1,n]
            c0 = fmaf(t, cov_w[n],       c0);
            c1 = fmaf(t, cov_w[16 + n],  c1);
        }
        const int rem = pix % HW;
        out_mean[pix * 2 + 0] = (float)(rem % Wn) + m0;  // x (col) + ofs
        out_mean[pix * 2 + 1] = (float)(rem / Wn) + m1;  // y (row) + ofs
        cvraw[pix * 2 + 0] = c0;
        cvraw[pix * 2 + 1] = c1;
    }
}

// ---------------------------------------------------------------------------
// Stage 2: per-batch scalar mean / rstd over the 4608 cv values.
// Deterministic LDS tree reduction (no float atomics -> replay-stable).
// ---------------------------------------------------------------------------
__global__ void k_stats(const float* __restrict__ cvraw,
                        float* __restrict__ stats)  // (Bn,2): mu, rstd
{
    __shared__ float s1[256];
    __shared__ float s2[256];
    const int b = blockIdx.x, t = threadIdx.x;
    const float* p = cvraw + (size_t)b * (2 * HW);

    float sum = 0.f, ssq = 0.f;
    for (int i = t; i < 2 * HW; i += 256) {
        const float v = p[i];
        sum += v;
        ssq = fmaf(v, v, ssq);
    }
    s1[t] = sum; s2[t] = ssq;
    __syncthreads();
    for (int off = 128; off > 0; off >>= 1) {
        if (t < off) { s1[t] += s1[t + off]; s2[t] += s2[t + off]; }
        __syncthreads();
    }
    if (t == 0) {
        const float n = (float)(2 * HW);
        const float mu = s1[0] / n;
        const float var = s2[0] / n - mu * mu + EPS_F;
        stats[b * 2 + 0] = mu;
        stats[b * 2 + 1] = rsqrtf(var);
    }
}

// ---------------------------------------------------------------------------
// Stage 3: streaming pass over corr. One block per pixel; 192 threads each
// move 3 float4s of the 2304-element (h2,w2) slice. corr1 = corr * factor,
// factor = 1 + exp(-0.5*(dy^2/cy + dx^2/cx)) / (6.28*sqrt(det)) inside the
// RADIUS window, else 1. NT hints keep the 340MB stream out of the caches.
// ---------------------------------------------------------------------------
__global__ void k_mask(const float* __restrict__ corr,
                       const float* __restrict__ cvraw,
                       const float* __restrict__ stats,
                       const float* __restrict__ out_mean,
                       float* __restrict__ out_corr,
                       float* __restrict__ out_det)
{
    __shared__ float prm[5];  // mean_x, mean_y, 0.5/cx, 0.5/cy, invden
    const int pix = blockIdx.x;
    const int t = threadIdx.x;

    if (t == 0) {
        const int bb = pix / HW;
        const float mu = stats[bb * 2 + 0];
        const float rs = stats[bb * 2 + 1];
        const float z0 = (cvraw[pix * 2 + 0] - mu) * rs;
        const float z1 = (cvraw[pix * 2 + 1] - mu) * rs;
        const float c0 = 5.0f / (1.0f + __expf(-z0)) + 0.05f;  // cov x
        const float c1 = 5.0f / (1.0f + __expf(-z1)) + 0.05f;  // cov y
        const float det = c0 * c1;
        out_det[pix] = det;
        prm[0] = out_mean[pix * 2 + 0];
        prm[1] = out_mean[pix * 2 + 1];
        prm[2] = 0.5f / c0;
        prm[3] = 0.5f / c1;
        prm[4] = 1.0f / (6.28f * sqrtf(det));
    }
    __syncthreads();

    const float mx = prm[0], my = prm[1];
    const float hx = prm[2], hy = prm[3], invden = prm[4];

    const f4* __restrict__ cin = (const f4*)(corr + (size_t)pix * HW);
    f4* __restrict__ cout = (f4*)(out_corr + (size_t)pix * HW);

#pragma unroll
    for (int i = 0; i < 3; ++i) {
        const int e = t + i * 192;        // float4 index, 0..575
        const int h2 = e / 12;            // 48/4 = 12 f4 per row
        const int w2 = (e % 12) * 4;

        f4 v = __builtin_nontemporal_load(cin + e);

        const float dy = (float)h2 - my;
        const float ey = dy * dy * hy;
        const bool oky = __builtin_fabsf(dy) <= RADIUS_F;
        const float dx0 = (float)w2 - mx;
#pragma unroll
        for (int j = 0; j < 4; ++j) {
            const float dx = dx0 + (float)j;
            float f = 1.0f;
            if (oky && __builtin_fabsf(dx) <= RADIUS_F)
                f = fmaf(__expf(-(ey + dx * dx * hx)), invden, 1.0f);
            v[j] *= f;
        }
        __builtin_nontemporal_store(v, cout + e);
    }
}

// ---------------------------------------------------------------------------
extern "C" void kernel_launch(void* const* d_in, const int* in_sizes, int n_in,
                              void* d_out, int out_size, void* d_ws, size_t ws_size,
                              hipStream_t stream) {
    (void)in_sizes; (void)n_in; (void)out_size; (void)ws_size;

    const float* x      = (const float*)d_in[0];
    const float* corr   = (const float*)d_in[1];
    const float* map_w  = (const float*)d_in[2];
    const float* map_b  = (const float*)d_in[3];
    const float* mean_w = (const float*)d_in[4];
    const float* mean_b = (const float*)d_in[5];
    const float* cov_w  = (const float*)d_in[6];
    const float* cov_b  = (const float*)d_in[7];

    float* out       = (float*)d_out;
    float* out_corr  = out;                         // (B,H,W,H,W)
    float* out_mean  = out + CORR_ELEMS;            // (B,H,W,2)
    float* out_det   = out_mean + (size_t)2 * NPIX; // (B,H*W)

    float* ws    = (float*)d_ws;
    float* cvraw = ws;              // 2*NPIX floats
    float* stats = ws + 2 * NPIX;   // 2*Bn floats

    k_mlp_wmma<<<NPIX / 16, 32, 0, stream>>>(x, map_w, map_b, mean_w, mean_b,
                                             cov_w, cov_b, out_mean, cvraw);
    k_stats<<<Bn, 256, 0, stream>>>(cvraw, stats);
    k_mask<<<NPIX, 192, 0, stream>>>(corr, cvraw, stats, out_mean,
                                     out_corr, out_det);
}